// MaxLayer_41077067219108
// MI455X (gfx1250) — compile-verified
//
#include <hip/hip_runtime.h>

typedef __attribute__((ext_vector_type(2))) float v2f;
typedef __attribute__((ext_vector_type(8))) float v8f;

#define BATCH   8
#define NDIM    2048
#define FDIM    64
#define KC      128                 // K-chunk staged in LDS
#define PSTR    160                 // floats per K-pair row (128 data + 32 pad): conflict-free b64 reads
#define WAVES   4
#define TPB     (WAVES * 32)

__global__ __launch_bounds__(TPB) void maxlayer_bmm_wmma(
    const float* __restrict__ a,   // [B, N, N]
    const float* __restrict__ x,   // [B, N, F]
    float* __restrict__ out)       // [B, N, F]
{
    // K-pair interleaved X tile: ldsX[(k/2)*PSTR + c*2 + (k&1)]
    __shared__ float ldsX[(KC / 2) * PSTR];   // 40 KB

    const int wg   = blockIdx.x;           // 0..255
    const int b    = wg >> 5;              // batch (32 workgroups per batch)
    const int rg   = wg & 31;              // 64-row group within batch
    const int tid  = threadIdx.x;
    const int wave = tid >> 5;
    const int lane = tid & 31;

    const int mlane = lane & 15;           // M (A) / N (B,C) index within tile
    const int khalf = lane >> 4;           // which K-pair this half-wave owns

    const int rowBase = rg * 64 + wave * 16;
    const float* aLane = a + ((size_t)b * NDIM + rowBase + mlane) * NDIM + khalf * 2;
    const float* xB    = x + (size_t)b * NDIM * FDIM;

    // per-lane LDS base for B fragments (advances by (kk/2)*PSTR per K-step)
    const float* bBase = &ldsX[khalf * PSTR + mlane * 2];

    v8f acc0 = {}, acc1 = {}, acc2 = {}, acc3 = {};

    for (int k0 = 0; k0 < NDIM; k0 += KC) {
        // ---- stage X[k0:k0+KC, 0:64] into K-pair-interleaved LDS ----
        // Each item: read float4 from two adjacent K-rows, emit interleaved
        // pairs as two aligned b128 LDS stores.
        for (int q = tid; q < (KC / 2) * (FDIM / 4); q += TPB) {
            const int p  = q >> 4;               // K-pair index within chunk
            const int c4 = (q & 15) * 4;         // column quad
            const float4 r0 = *(const float4*)(xB + (size_t)(k0 + 2 * p    ) * FDIM + c4);
            const float4 r1 = *(const float4*)(xB + (size_t)(k0 + 2 * p + 1) * FDIM + c4);
            const float4 lo = { r0.x, r1.x, r0.y, r1.y };
            const float4 hi = { r0.z, r1.z, r0.w, r1.w };
            float* d = &ldsX[p * PSTR + c4 * 2];
            *(float4*)(d)     = lo;
            *(float4*)(d + 4) = hi;
        }
        __syncthreads();

        const float* aPtr = aLane + k0;
        if (k0 + KC < NDIM)                       // hint next A chunk for this lane's row
            __builtin_prefetch(aPtr + KC, 0, 1);

        // ---- K loop over the chunk, 4 deep per WMMA ----
        #pragma unroll 4
        for (int kk = 0; kk < KC; kk += 4) {
            // A fragment (16x4 f32): lane holds A[row][kb], A[row][kb+1] - one b64
            const v2f afrag = *(const v2f*)(aPtr + kk);

            // B fragments (4x16 f32): one ds_load_b64 each, tile columns are imm offsets
            const float* bp = bBase + (kk >> 1) * PSTR;
            const v2f b0 = *(const v2f*)(bp +  0);
            const v2f b1 = *(const v2f*)(bp + 32);
            const v2f b2 = *(const v2f*)(bp + 64);
            const v2f b3 = *(const v2f*)(bp + 96);

            acc0 = __builtin_amdgcn_wmma_f32_16x16x4_f32(false, afrag, false, b0,
                                                         (short)0, acc0, false, false);
            acc1 = __builtin_amdgcn_wmma_f32_16x16x4_f32(false, afrag, false, b1,
                                                         (short)0, acc1, false, false);
            acc2 = __builtin_amdgcn_wmma_f32_16x16x4_f32(false, afrag, false, b2,
                                                         (short)0, acc2, false, false);
            acc3 = __builtin_amdgcn_wmma_f32_16x16x4_f32(false, afrag, false, b3,
                                                         (short)0, acc3, false, false);
        }
        __syncthreads();
    }

    // ---- fused threshold + store ----
    // C/D layout: VGPR r -> M = r + khalf*8, N = mlane (+16 per tile column)
    const int row0 = rowBase + khalf * 8;
    float* outBase = out + ((size_t)b * NDIM + row0) * FDIM;
    #pragma unroll
    for (int r = 0; r < 8; ++r) {
        float* o = outBase + (size_t)r * FDIM;
        o[ 0 + mlane] = acc0[r] > 0.5f ? 1.0f : 0.0f;
        o[16 + mlane] = acc1[r] > 0.5f ? 1.0f : 0.0f;
        o[32 + mlane] = acc2[r] > 0.5f ? 1.0f : 0.0f;
        o[48 + mlane] = acc3[r] > 0.5f ? 1.0f : 0.0f;
    }
}

extern "C" void kernel_launch(void* const* d_in, const int* in_sizes, int n_in,
                              void* d_out, int out_size, void* d_ws, size_t ws_size,
                              hipStream_t stream) {
    (void)in_sizes; (void)n_in; (void)d_ws; (void)ws_size; (void)out_size;
    const float* x = (const float*)d_in[0];   // [8, 2048, 64]
    const float* a = (const float*)d_in[1];   // [8, 2048, 2048]
    float* out = (float*)d_out;               // [8, 2048, 64]

    const int grid = BATCH * (NDIM / 64);     // 256 workgroups, 4 waves each
    maxlayer_bmm_wmma<<<grid, TPB, 0, stream>>>(a, x, out);
}